// SSAMLayer_53137335386832
// MI455X (gfx1250) — compile-verified
//
#include <hip/hip_runtime.h>
#include <math.h>

typedef __attribute__((ext_vector_type(2))) float v2f;
typedef __attribute__((ext_vector_type(8))) float v8f;

#define Cdim 256
#define PIX  4096              // H*W = 64*64
#define IMG  (Cdim * PIX)      // elements per (b,band) image block

// ---------------------------------------------------------------------------
// CDNA5 fp32 WMMA: D(16x16) = A(16x4) * B(4x16) + C
// A layout: lane L -> row M = L&15, VGPR j -> K = 2*(L>>4)+j
// B layout: lane L -> col N = L&15, VGPR j -> K = 2*(L>>4)+j
// D layout: lane L -> col N = L&15, VGPR v -> row M = v + 8*(L>>4)
// ---------------------------------------------------------------------------
__device__ __forceinline__ v8f wmma_f32(v2f a, v2f b, v8f c) {
  return __builtin_amdgcn_wmma_f32_16x16x4_f32(false, a, false, b, (short)0, c,
                                               false, false);
}

// ---------------------------------------------------------------------------
// Band projection matrices (transposed): Mt[band][m][n] = M_band[n][m]
// M[n][m] = (1/256)*(cos(2*pi*s*m/256) + 2*sum_{j=1..J-1} cos(2*pi*((s+j)m - j n)/256))
// (irfft re-indexes the sliced spectrum to start at frequency 0; DC imag dropped)
// ---------------------------------------------------------------------------
__global__ void band_matrix_kernel(float* __restrict__ Mt) {
  const int m = threadIdx.x;
  const int n = blockIdx.x;
  const int band = blockIdx.y;
  const int s = band * 32;
  const int J = (band == 3) ? 33 : 32;
  const float step = 6.28318530717958647692f / 256.0f;
  float acc = cosf((float)((s * m) & 255) * step);
  for (int j = 1; j < J; ++j) {
    int t = ((s + j) * m - j * n) % 256;
    if (t < 0) t += 256;
    acc += 2.0f * cosf((float)t * step);
  }
  Mt[band * 65536 + m * 256 + n] = acc * (1.0f / 256.0f);
}

// w_eff[band][c] = sum_a spline_w[band][a][c] * scalers[band][a][c]
__global__ void weff_kernel(const float* __restrict__ spw,
                            const float* __restrict__ scal,
                            float* __restrict__ weff) {
  const int band = blockIdx.x, c = threadIdx.x;
  const float* a = spw + band * 65536;
  const float* b = scal + band * 65536;
  float acc = 0.0f;
  for (int r = 0; r < 256; ++r) acc += a[r * 256 + c] * b[r * 256 + c];
  weff[band * 256 + c] = acc;
}

// Wp[tap][ci][co] = W[co][ci][tap]  (coalesced A-fragment loads in the conv)
__global__ void repack_kernel(const float* __restrict__ w, float* __restrict__ wp) {
  const int idx = blockIdx.x * 256 + threadIdx.x;   // exactly 9*256*256
  const int tap = idx >> 16;
  const int rem = idx & 65535;
  const int ci = rem >> 8, co = rem & 255;
  wp[idx] = w[(co * 256 + ci) * 9 + tap];
}

// ---------------------------------------------------------------------------
// Band GEMM: Y[(b,band)][p][co] = sum_c X[b][c][p] * M_band[co][c]
// Block = 128 threads = 4 waves; wave computes 16p x 64co (4 accumulators).
// ---------------------------------------------------------------------------
__global__ __launch_bounds__(128) void band_gemm_kernel(
    const float* __restrict__ x, const float* __restrict__ Mt,
    float* __restrict__ Y) {
  const int tid = threadIdx.x;
  const int wv = tid >> 5, lane = tid & 31;
  const int row = lane & 15, half = lane >> 4;
  const int pt = blockIdx.x & 255;
  const int bi = blockIdx.x >> 8;          // (b*4 + band)
  const int p0 = pt * 16;
  const int co0 = wv * 64;
  const float* Xb = x + (size_t)(bi >> 2) * IMG;     // NCHW image of batch b
  const float* Mb = Mt + (size_t)(bi & 3) * 65536;   // [c][co]

  v8f acc[4];
  const v8f vzero = {0.f, 0.f, 0.f, 0.f, 0.f, 0.f, 0.f, 0.f};
#pragma unroll
  for (int t = 0; t < 4; ++t) acc[t] = vzero;

  for (int k0 = 0; k0 < 256; k0 += 4) {
    const int k = k0 + 2 * half;
    v2f a;
    a.x = Xb[k * PIX + p0 + row];
    a.y = Xb[(k + 1) * PIX + p0 + row];
    const float* mrow = Mb + k * 256;
#pragma unroll
    for (int t = 0; t < 4; ++t) {
      v2f b;
      b.x = mrow[co0 + t * 16 + row];
      b.y = mrow[256 + co0 + t * 16 + row];
      acc[t] = wmma_f32(a, b, acc[t]);
    }
  }

  float* yb = Y + (size_t)bi * IMG + (size_t)p0 * 256;
#pragma unroll
  for (int t = 0; t < 4; ++t)
#pragma unroll
    for (int v = 0; v < 8; ++v) {
      const int p = v + 8 * half;
      yb[p * 256 + co0 + t * 16 + row] = acc[t][v];
    }
}

// ---------------------------------------------------------------------------
// conv3x3 256->256 (pad 1) as implicit GEMM over K = 9 taps x 256 ci.
// Block = 256 threads = 8 waves, computes one output row h, 128 co, 64 w.
// Wave: 32co x 32w tile = 4 WMMA accumulators.
// fuse_bn_relu=1: y = relu(bn(conv+b));  =0: y = conv+b
// ---------------------------------------------------------------------------
__global__ __launch_bounds__(256) void conv3x3_kernel(
    const float* __restrict__ in, float* __restrict__ out,
    const float* __restrict__ Wp, const float* __restrict__ bias,
    const float* __restrict__ bng, const float* __restrict__ bnb,
    const float* __restrict__ bnm, const float* __restrict__ bnv,
    int fuse_bn_relu) {
  __shared__ float smem[4 * 208];   // [ci(4), stride 208][r(3)*66 + x(66)], padded
  const int tid = threadIdx.x;
  const int h = blockIdx.x, cog = blockIdx.y, n = blockIdx.z;
  const int wv = tid >> 5, lane = tid & 31, row = lane & 15, half = lane >> 4;
  const int co_base = cog * 128 + (wv >> 1) * 32;
  const int wbase = (wv & 1) * 32;
  const float* I = in + (size_t)n * IMG;

  v8f acc[2][2];
  const v8f vzero = {0.f, 0.f, 0.f, 0.f, 0.f, 0.f, 0.f, 0.f};
  acc[0][0] = vzero; acc[0][1] = vzero; acc[1][0] = vzero; acc[1][1] = vzero;

  for (int ci0 = 0; ci0 < 256; ci0 += 4) {
    __syncthreads();
    // stage 4 input channels x 3 rows x (64+2 halo) into LDS (zero-padded)
    for (int idx = tid; idx < 792; idx += 256) {
      const int cil = idx / 198;
      const int rem = idx - cil * 198;
      const int r = rem / 66;
      const int xx = rem - r * 66;
      const int hin = h + r - 1, win = xx - 1;
      float v = 0.0f;
      if ((unsigned)hin < 64u && (unsigned)win < 64u) {
        v = I[(ci0 + cil) * PIX + hin * 64 + win];
        if (ci0 < 252)
          __builtin_prefetch(&I[(ci0 + 4 + cil) * PIX + hin * 64 + win], 0, 0);
      }
      smem[cil * 208 + r * 66 + xx] = v;
    }
    __syncthreads();

#pragma unroll
    for (int r = 0; r < 3; ++r)
#pragma unroll
      for (int s = 0; s < 3; ++s) {
        const float* wp = Wp + (r * 3 + s) * 65536 + (ci0 + 2 * half) * 256;
        v2f afr[2], bfr[2];
        afr[0].x = wp[co_base + row];
        afr[0].y = wp[256 + co_base + row];
        afr[1].x = wp[co_base + 16 + row];
        afr[1].y = wp[256 + co_base + 16 + row];
        const float* sm = smem + (2 * half) * 208 + r * 66 + wbase + row + s;
        bfr[0].x = sm[0];   bfr[0].y = sm[208];
        bfr[1].x = sm[16];  bfr[1].y = sm[16 + 208];
        acc[0][0] = wmma_f32(afr[0], bfr[0], acc[0][0]);
        acc[0][1] = wmma_f32(afr[0], bfr[1], acc[0][1]);
        acc[1][0] = wmma_f32(afr[1], bfr[0], acc[1][0]);
        acc[1][1] = wmma_f32(afr[1], bfr[1], acc[1][1]);
      }
  }

  float* O = out + (size_t)n * IMG;
#pragma unroll
  for (int ct = 0; ct < 2; ++ct)
#pragma unroll
    for (int wt = 0; wt < 2; ++wt)
#pragma unroll
      for (int v = 0; v < 8; ++v) {
        const int co = co_base + ct * 16 + v + 8 * half;
        const int w = wbase + wt * 16 + row;
        float val = acc[ct][wt][v];
        if (fuse_bn_relu) {
          val = (val + bias[co] - bnm[co]) * (bng[co] * rsqrtf(bnv[co] + 1e-5f))
              + bnb[co];
          val = fmaxf(val, 0.0f);
        } else {
          val += bias[co];
        }
        O[co * PIX + h * 64 + w] = val;
      }
}

// Global average pool: g[n*256+c] = mean_p T[n][c][p]
__global__ void gap_kernel(const float* __restrict__ T, float* __restrict__ g) {
  __shared__ float red[256];
  const int nc = blockIdx.x;
  const float* base = T + (size_t)(nc >> 8) * IMG + (size_t)(nc & 255) * PIX;
  float s = 0.0f;
  for (int p = threadIdx.x; p < 4096; p += 256) s += base[p];
  red[threadIdx.x] = s;
  __syncthreads();
  for (int off = 128; off > 0; off >>= 1) {
    if (threadIdx.x < off) red[threadIdx.x] += red[threadIdx.x + off];
    __syncthreads();
  }
  if (threadIdx.x == 0) g[nc] = red[0] * (1.0f / 4096.0f);
}

// Channel attention: cw = sigmoid(W2 * gelu(W1*g + b1) + b2)
__global__ void ca_kernel(const float* __restrict__ g,
                          const float* __restrict__ w1, const float* __restrict__ b1,
                          const float* __restrict__ w2, const float* __restrict__ b2,
                          float* __restrict__ cw) {
  __shared__ float gs[256];
  __shared__ float h1[64];
  const int n = blockIdx.x, t = threadIdx.x;
  gs[t] = g[n * 256 + t];
  __syncthreads();
  if (t < 64) {
    float a = b1[t];
    for (int c = 0; c < 256; ++c) a += w1[t * 256 + c] * gs[c];
    h1[t] = 0.5f * a * (1.0f + erff(a * 0.70710678118654752f));  // exact GELU
  }
  __syncthreads();
  float a = b2[t];
  for (int j = 0; j < 64; ++j) a += w2[t * 64 + j] * h1[j];
  cw[n * 256 + t] = 1.0f / (1.0f + expf(-a));
}

// t *= cw (in place) and per-pixel channel mean/max -> S2[n][2][4096]
__global__ void scale_stats_kernel(float* __restrict__ T,
                                   const float* __restrict__ cw,
                                   float* __restrict__ S2) {
  const int n = blockIdx.y;
  const int h = blockIdx.x * 4 + (threadIdx.x >> 6);
  const int w = threadIdx.x & 63;
  const int p = h * 64 + w;
  float* base = T + (size_t)n * IMG + p;
  const float* cwn = cw + n * 256;
  float s = 0.0f, mx = -3.402823466e+38f;
  for (int c = 0; c < 256; ++c) {
    float v = base[c * PIX] * cwn[c];
    base[c * PIX] = v;
    s += v;
    mx = fmaxf(mx, v);
  }
  S2[n * 8192 + p] = s * (1.0f / 256.0f);
  S2[n * 8192 + 4096 + p] = mx;
}

// Spatial attention: sw = sigmoid(conv3x3_{2->1}([avg;max]))
__global__ void sa_kernel(const float* __restrict__ S2,
                          const float* __restrict__ saw,
                          const float* __restrict__ sab,
                          float* __restrict__ sw_) {
  const int gidx = blockIdx.x * 256 + threadIdx.x;  // 16*4096
  const int n = gidx >> 12, p = gidx & 4095;
  const int h = p >> 6, w = p & 63;
  float acc = sab[0];
#pragma unroll
  for (int ch = 0; ch < 2; ++ch)
#pragma unroll
    for (int r = 0; r < 3; ++r)
#pragma unroll
      for (int s = 0; s < 3; ++s) {
        const int hin = h + r - 1, win = w + s - 1;
        if ((unsigned)hin < 64u && (unsigned)win < 64u)
          acc += saw[ch * 9 + r * 3 + s] * S2[n * 8192 + ch * 4096 + hin * 64 + win];
      }
  sw_[gidx] = 1.0f / (1.0f + expf(-acc));
}

// Final: out[b][c][h][w] = 0.25 * sum_i weff[i][c] * T'[(b,i) raw (H,W,C) elem] * sw
// T' is read in its natural (C,H,W)-reinterpreted flat layout: f = h*16384+w*256+c,
// with sw applied at NCHW pixel pp = f mod 4096 = (w%16)*256 + c.
__global__ __launch_bounds__(256) void final_kernel(
    const float* __restrict__ T, const float* __restrict__ sw_,
    const float* __restrict__ weff, float* __restrict__ out) {
  __shared__ float ls[16 * 257];
  const int w0 = blockIdx.x * 16, h = blockIdx.y, b = blockIdx.z;
  const int c = threadIdx.x;
  float accs[16];
#pragma unroll
  for (int k = 0; k < 16; ++k) accs[k] = 0.0f;

  for (int i = 0; i < 4; ++i) {
    const int n = b * 4 + i;
    const float* Tn = T + (size_t)n * IMG + h * 16384 + w0 * 256 + c;
    const float* swn = sw_ + n * 4096;
    const float we = weff[i * 256 + c] * 0.25f;
#pragma unroll
    for (int k = 0; k < 16; ++k)
      accs[k] += we * Tn[k * 256] * swn[(k << 8) + c];   // w0%16==0 -> (w%16)==k
  }
#pragma unroll
  for (int k = 0; k < 16; ++k) ls[k * 257 + c] = accs[k];
  __syncthreads();

  float* ob = out + (size_t)b * IMG + h * 64 + w0;
#pragma unroll
  for (int j = 0; j < 16; ++j) {
    const int e = c + 256 * j;
    const int cc = e >> 4, wl = e & 15;
    ob[cc * PIX + wl] = ls[wl * 257 + cc];
  }
}

// ---------------------------------------------------------------------------
extern "C" void kernel_launch(void* const* d_in, const int* in_sizes, int n_in,
                              void* d_out, int out_size, void* d_ws, size_t ws_size,
                              hipStream_t stream) {
  (void)in_sizes; (void)n_in; (void)out_size; (void)ws_size;
  const float* x    = (const float*)d_in[0];
  const float* spw  = (const float*)d_in[1];
  const float* scal = (const float*)d_in[2];
  const float* ftw1 = (const float*)d_in[3];
  const float* ftb1 = (const float*)d_in[4];
  const float* bng  = (const float*)d_in[5];
  const float* bnb  = (const float*)d_in[6];
  const float* bnm  = (const float*)d_in[7];
  const float* bnv  = (const float*)d_in[8];
  const float* ftw2 = (const float*)d_in[9];
  const float* ftb2 = (const float*)d_in[10];
  const float* caw1 = (const float*)d_in[11];
  const float* cab1 = (const float*)d_in[12];
  const float* caw2 = (const float*)d_in[13];
  const float* cab2 = (const float*)d_in[14];
  const float* saw  = (const float*)d_in[15];
  const float* sab  = (const float*)d_in[16];
  float* out = (float*)d_out;

  float* ws = (float*)d_ws;
  float* Mt   = ws;  ws += 262144;     // 4 x 256 x 256 (transposed band mats)
  float* Wp1  = ws;  ws += 589824;     // 9 x 256 x 256
  float* Wp2  = ws;  ws += 589824;
  float* weff = ws;  ws += 1024;
  float* g    = ws;  ws += 4096;
  float* cw   = ws;  ws += 4096;
  float* S2   = ws;  ws += 131072;     // 16 x 2 x 4096
  float* swb  = ws;  ws += 65536;      // 16 x 4096
  float* buf0 = ws;  ws += 16777216;   // 16 x 256 x 4096 each
  float* buf1 = ws;  ws += 16777216;
  float* buf2 = ws;

  band_matrix_kernel<<<dim3(256, 4), 256, 0, stream>>>(Mt);
  weff_kernel<<<4, 256, 0, stream>>>(spw, scal, weff);
  repack_kernel<<<2304, 256, 0, stream>>>(ftw1, Wp1);
  repack_kernel<<<2304, 256, 0, stream>>>(ftw2, Wp2);

  // band decompose (WMMA GEMM): x -> buf0 = (B,NB,H,W,C) raw == (16,C,H,W) NCHW
  band_gemm_kernel<<<4096, 128, 0, stream>>>(x, Mt, buf0);

  // feature_transform #1: conv+BN+ReLU, conv+bias
  conv3x3_kernel<<<dim3(64, 2, 16), 256, 0, stream>>>(buf0, buf1, Wp1, ftb1,
                                                      bng, bnb, bnm, bnv, 1);
  conv3x3_kernel<<<dim3(64, 2, 16), 256, 0, stream>>>(buf1, buf2, Wp2, ftb2,
                                                      bng, bnb, bnm, bnv, 0);
  // feature_transform #2 (reshape round-trip is a memory no-op)
  conv3x3_kernel<<<dim3(64, 2, 16), 256, 0, stream>>>(buf2, buf0, Wp1, ftb1,
                                                      bng, bnb, bnm, bnv, 1);
  conv3x3_kernel<<<dim3(64, 2, 16), 256, 0, stream>>>(buf0, buf1, Wp2, ftb2,
                                                      bng, bnb, bnm, bnv, 0);  // T

  gap_kernel<<<4096, 256, 0, stream>>>(buf1, g);
  ca_kernel<<<16, 256, 0, stream>>>(g, caw1, cab1, caw2, cab2, cw);
  scale_stats_kernel<<<dim3(16, 16), 256, 0, stream>>>(buf1, cw, S2);
  sa_kernel<<<256, 256, 0, stream>>>(S2, saw, sab, swb);
  final_kernel<<<dim3(4, 64, 4), 256, 0, stream>>>(buf1, swb, weff, out);
}